// Head_1949915152919
// MI455X (gfx1250) — compile-verified
//
#include <hip/hip_runtime.h>

// Problem constants (match reference)
constexpr int B_ = 8;
constexpr int T_ = 2048;
constexpr int C_ = 1024;
constexpr int H_ = 64;
constexpr int NT = T_ / 16;          // 128 row/col tiles
constexpr float SQRT_C = 32.0f;      // C**0.5

typedef float v2f __attribute__((ext_vector_type(2)));
typedef float v8f __attribute__((ext_vector_type(8)));

static __device__ __forceinline__ v8f wmma_f32(v2f a, v2f b, v8f c) {
  // D = A(16x4 f32) * B(4x16 f32) + C(16x16 f32)
  return __builtin_amdgcn_wmma_f32_16x16x4_f32(
      /*neg_a=*/false, a, /*neg_b=*/false, b,
      /*c_mod=*/(short)0, c, /*reuse_a=*/false, /*reuse_b=*/false);
}

static __device__ __forceinline__ v8f zero8() {
  v8f z = {0.f, 0.f, 0.f, 0.f, 0.f, 0.f, 0.f, 0.f};
  return z;
}

// -------------------------------------------------------------------------
// Kernel 1: projections.  OUT[z][b, t, :] = X[b, t, :] @ W_z   (z = k,q,v)
// One wave computes a 16x64 output tile (4 accumulators of 16x16).
// grid = (NT, B, 3), block = 32
// -------------------------------------------------------------------------
__global__ __launch_bounds__(32) void proj_kernel(
    const float* __restrict__ X, const float* __restrict__ Wk,
    const float* __restrict__ Wq, const float* __restrict__ Wv,
    float* __restrict__ KQV) {
  const int i = blockIdx.x;            // row tile
  const int b = blockIdx.y;
  const int z = blockIdx.z;
  const float* W = (z == 0) ? Wk : (z == 1) ? Wq : Wv;
  float* O = KQV + (size_t)z * B_ * T_ * H_;

  const int lane = threadIdx.x;
  const int half = lane >> 4;          // 0: K/M low, 1: K/M high
  const int lm = lane & 15;

  const float* xrow = X + ((size_t)b * T_ + i * 16 + lm) * C_;

  v8f c0 = zero8(), c1 = zero8(), c2 = zero8(), c3 = zero8();

  for (int k = 0; k < C_; k += 4) {
    const int kr = k + 2 * half;
    // A fragment: A[M=lm][K=kr], A[M=lm][K=kr+1]  (contiguous -> b64 load)
    v2f a = *(const v2f*)(xrow + kr);
    // B fragments: B[K][N] = W[kr][n]  (W is [C,H] row-major)
    const float* wr0 = W + (size_t)kr * H_;
    const float* wr1 = wr0 + H_;
    v2f b0 = {wr0[lm], wr1[lm]};
    v2f b1 = {wr0[16 + lm], wr1[16 + lm]};
    v2f b2 = {wr0[32 + lm], wr1[32 + lm]};
    v2f b3 = {wr0[48 + lm], wr1[48 + lm]};
    c0 = wmma_f32(a, b0, c0);
    c1 = wmma_f32(a, b1, c1);
    c2 = wmma_f32(a, b2, c2);
    c3 = wmma_f32(a, b3, c3);
  }

  float* orow = O + ((size_t)b * T_ + i * 16) * H_;
#pragma unroll
  for (int r = 0; r < 8; ++r) {
    const int M = r + 8 * half;
    orow[(size_t)M * H_ + lm]      = c0[r];
    orow[(size_t)M * H_ + 16 + lm] = c1[r];
    orow[(size_t)M * H_ + 32 + lm] = c2[r];
    orow[(size_t)M * H_ + 48 + lm] = c3[r];
  }
}

// -------------------------------------------------------------------------
// Kernel 2: per-COLUMN softmax stats (softmax over axis=1 / the t axis).
// For column s: m_s = max_{t>=s} (k_t . q_s)*32 ; d_s = sum exp(. - m_s).
// One wave owns one 16-column tile; iterates row tiles i = j..NT-1 with an
// online (m,d) rescaled reduction. Score tiles via WMMA.
// grid = (NT, B), block = 32
// -------------------------------------------------------------------------
__global__ __launch_bounds__(32) void colstats_kernel(
    const float* __restrict__ Kbuf, const float* __restrict__ Qbuf,
    float* __restrict__ mbuf, float* __restrict__ dinvbuf) {
  const int j = blockIdx.x;            // column (s) tile
  const int b = blockIdx.y;
  const int lane = threadIdx.x;
  const int half = lane >> 4;
  const int lm = lane & 15;
  const int s = j * 16 + lm;           // column handled by this lane

  const float* qrow = Qbuf + ((size_t)b * T_ + s) * H_;

  float m_run = -INFINITY;
  float d_run = 0.0f;

  for (int i = j; i < NT; ++i) {
    const float* krow = Kbuf + ((size_t)b * T_ + i * 16 + lm) * H_;
    __builtin_prefetch(krow + 16 * H_, 0, 3);  // next row tile

    v8f c = zero8();
#pragma unroll
    for (int k = 0; k < H_; k += 4) {
      const int kr = k + 2 * half;
      v2f a  = *(const v2f*)(krow + kr);   // A[M=t_local][K=h]
      v2f bb = *(const v2f*)(qrow + kr);   // B[K=h][N=s_local] = Q[s][h]
      c = wmma_f32(a, bb, c);
    }

    float vals[8];
    float tmax = -INFINITY;
#pragma unroll
    for (int r = 0; r < 8; ++r) {
      const int t = i * 16 + r + 8 * half;
      const float sv = (t >= s) ? c[r] * SQRT_C : -INFINITY;
      vals[r] = sv;
      tmax = fmaxf(tmax, sv);
    }
    tmax = fmaxf(tmax, __shfl_xor(tmax, 16));   // combine lane halves
    const float newm = fmaxf(m_run, tmax);      // finite after first tile

    float lsum = 0.0f;
#pragma unroll
    for (int r = 0; r < 8; ++r) lsum += expf(vals[r] - newm);
    lsum += __shfl_xor(lsum, 16);

    d_run = d_run * expf(m_run - newm) + lsum;
    m_run = newm;
  }

  if (half == 0) {
    mbuf[(size_t)b * T_ + s] = m_run;
    dinvbuf[(size_t)b * T_ + s] = 1.0f / d_run;
  }
}

// -------------------------------------------------------------------------
// Kernel 3: output.  out[t,:] = sum_{s<=t} exp(score[t,s]-m_s) * (v_s * d_s^-1)
// One wave owns one 16-row (t) tile x full H=64.  Per s-tile:
//   S tile (WMMA f32) -> P = exp(S*32 - m_s) (masked) -> LDS transpose ->
//   OUT += P @ V'  (WMMA f32, V' = V * dinv folded on load).
// grid = (NT, B), block = 32
// -------------------------------------------------------------------------
__global__ __launch_bounds__(32) void attnout_kernel(
    const float* __restrict__ Kbuf, const float* __restrict__ Qbuf,
    const float* __restrict__ Vbuf, const float* __restrict__ mbuf,
    const float* __restrict__ dinvbuf, float* __restrict__ out) {
  const int i = blockIdx.x;            // row (t) tile
  const int b = blockIdx.y;
  const int lane = threadIdx.x;
  const int half = lane >> 4;
  const int lm = lane & 15;

  __shared__ float P[16][17];          // padded to dodge bank conflicts

  const float* krow = Kbuf + ((size_t)b * T_ + i * 16 + lm) * H_;

  v8f o0 = zero8(), o1 = zero8(), o2 = zero8(), o3 = zero8();

  for (int j = 0; j <= i; ++j) {
    // ---- score tile S[t_local, s_local] ----
    const float* qrow = Qbuf + ((size_t)b * T_ + j * 16 + lm) * H_;
    v8f c = zero8();
#pragma unroll
    for (int k = 0; k < H_; k += 4) {
      const int kr = k + 2 * half;
      v2f a  = *(const v2f*)(krow + kr);
      v2f bb = *(const v2f*)(qrow + kr);
      c = wmma_f32(a, bb, c);
    }

    // ---- P = exp(S*32 - m_s), causal mask, stash to LDS ----
    const int s = j * 16 + lm;                 // this lane's column
    const float m_s = mbuf[(size_t)b * T_ + s];
#pragma unroll
    for (int r = 0; r < 8; ++r) {
      const int M = r + 8 * half;
      const int t = i * 16 + M;
      P[M][lm] = (t >= s) ? expf(c[r] * SQRT_C - m_s) : 0.0f;
    }
    __syncthreads();

    // ---- OUT += P(16x16) @ V'(16x64), V' = V * dinv ----
#pragma unroll
    for (int kk = 0; kk < 16; kk += 4) {
      const int kl = kk + 2 * half;            // local k (s) index pair base
      v2f a = {P[lm][kl], P[lm][kl + 1]};      // A[M=lm][K=kl..kl+1]

      const int srow = j * 16 + kl;
      const float di0 = dinvbuf[(size_t)b * T_ + srow];
      const float di1 = dinvbuf[(size_t)b * T_ + srow + 1];
      const float* v0 = Vbuf + ((size_t)b * T_ + srow) * H_;
      const float* v1 = v0 + H_;

      v2f b0 = {v0[lm] * di0,       v1[lm] * di1};
      v2f b1 = {v0[16 + lm] * di0,  v1[16 + lm] * di1};
      v2f b2 = {v0[32 + lm] * di0,  v1[32 + lm] * di1};
      v2f b3 = {v0[48 + lm] * di0,  v1[48 + lm] * di1};

      o0 = wmma_f32(a, b0, o0);
      o1 = wmma_f32(a, b1, o1);
      o2 = wmma_f32(a, b2, o2);
      o3 = wmma_f32(a, b3, o3);
    }
    __syncthreads();
  }

  float* orow = out + ((size_t)b * T_ + i * 16) * H_;
#pragma unroll
  for (int r = 0; r < 8; ++r) {
    const int M = r + 8 * half;
    orow[(size_t)M * H_ + lm]      = o0[r];
    orow[(size_t)M * H_ + 16 + lm] = o1[r];
    orow[(size_t)M * H_ + 32 + lm] = o2[r];
    orow[(size_t)M * H_ + 48 + lm] = o3[r];
  }
}

// -------------------------------------------------------------------------
// Host launcher
// -------------------------------------------------------------------------
extern "C" void kernel_launch(void* const* d_in, const int* in_sizes, int n_in,
                              void* d_out, int out_size, void* d_ws,
                              size_t ws_size, hipStream_t stream) {
  const float* X  = (const float*)d_in[0];   // [B,T,C]
  const float* Wk = (const float*)d_in[1];   // [C,H]
  const float* Wq = (const float*)d_in[2];   // [C,H]
  const float* Wv = (const float*)d_in[3];   // [C,H]
  float* out = (float*)d_out;                // [B,T,H]

  // Workspace layout (floats): K | Q | V | m | dinv
  const size_t BTH = (size_t)B_ * T_ * H_;
  float* ws   = (float*)d_ws;
  float* KQV  = ws;                 // 3 * BTH
  float* Kbuf = KQV;
  float* Qbuf = KQV + BTH;
  float* Vbuf = KQV + 2 * BTH;
  float* mbuf = KQV + 3 * BTH;      // B*T
  float* dinv = mbuf + (size_t)B_ * T_;

  proj_kernel<<<dim3(NT, B_, 3), 32, 0, stream>>>(X, Wk, Wq, Wv, KQV);
  colstats_kernel<<<dim3(NT, B_), 32, 0, stream>>>(Kbuf, Qbuf, mbuf, dinv);
  attnout_kernel<<<dim3(NT, B_), 32, 0, stream>>>(Kbuf, Qbuf, Vbuf, mbuf,
                                                  dinv, out);
}